// SeparableTensorProduct_41369124995686
// MI455X (gfx1250) — compile-verified
//
#include <hip/hip_runtime.h>

// SeparableTensorProduct on MI455X (gfx1250, wave32, WMMA, async LDS<->global).
// Fused single pass per 16-edge tile:
//   stage x/y via global_load_async_to_lds_b128 (ASYNCcnt)
//   -> phase 1: all-k Z panels from constexpr-folded Wigner-3j FMAs
//   -> phase 2: per-l channel mix on matrix pipes (V_WMMA_F32_16X16X4_F32)
//   -> drain output tile via global_store_async_from_lds_b128 (coalesced).

typedef float v2f __attribute__((ext_vector_type(2)));
typedef float v8f __attribute__((ext_vector_type(8)));

#define NDIM1 480   // x feature dim: 128*1 + 64*3 + 32*5
#define NDIM2 9     // y feature dim: 1 + 3 + 5
#define WNUM  960   // per-edge TP weights
#define TILE_E 16   // edges per block (= WMMA M)

// LDS pool layout (bytes)
#define OFF_SX 0                        // float sx[16][480]   = 30720 B
#define OFF_SY 30720                    // float sy[16][9]     =   576 B
#define OFF_SZ 31296                    // float sz[NK][16][FANIN+4]  (<= 113920 B)
#define OFF_SOUT (31296 + 5 * 16 * (352 + 4) * 4)        // 145216
#define LDS_BYTES (OFF_SOUT + TILE_E * 192 * 4)          // 157504 B (l=1 widest: 192 cols)

// ---- sparse real-basis Wigner-3j tables (a: x-comp, b: y-comp, c: out comp) ----
struct W3JEnt { int a, b, c; float v; };

__device__ constexpr W3JEnt ENT[120] = {
  // K000 [0,1)
  {0,0,0, 1.0f},
  // K011 [1,4)  delta/sqrt(3)
  {0,0,0, 0.57735027f},{0,1,1, 0.57735027f},{0,2,2, 0.57735027f},
  // K022 [4,9)  delta/sqrt(5)
  {0,0,0, 0.44721360f},{0,1,1, 0.44721360f},{0,2,2, 0.44721360f},{0,3,3, 0.44721360f},{0,4,4, 0.44721360f},
  // K101 [9,12)
  {0,0,0, 0.57735027f},{1,0,1, 0.57735027f},{2,0,2, 0.57735027f},
  // K110 [12,15)
  {0,0,0, 0.57735027f},{1,1,0, 0.57735027f},{2,2,0, 0.57735027f},
  // K111 [15,21)  Levi-Civita / sqrt(6)
  {0,1,2, 0.40824829f},{1,2,0, 0.40824829f},{2,0,1, 0.40824829f},
  {2,1,0,-0.40824829f},{1,0,2,-0.40824829f},{0,2,1,-0.40824829f},
  // K112 [21,32)  sym-traceless projection
  {0,2,0, 0.31622777f},{2,0,0, 0.31622777f},
  {0,1,1, 0.31622777f},{1,0,1, 0.31622777f},
  {0,0,2,-0.18257419f},{1,1,2, 0.36514837f},{2,2,2,-0.18257419f},
  {1,2,3, 0.31622777f},{2,1,3, 0.31622777f},
  {2,2,4, 0.31622777f},{0,0,4,-0.31622777f},
  // K121 [32,43)  (permutation of K112)
  {0,0,2, 0.31622777f},{2,0,0, 0.31622777f},
  {0,1,1, 0.31622777f},{1,1,0, 0.31622777f},
  {0,2,0,-0.18257419f},{1,2,1, 0.36514837f},{2,2,2,-0.18257419f},
  {1,3,2, 0.31622777f},{2,3,1, 0.31622777f},
  {2,4,2, 0.31622777f},{0,4,0,-0.31622777f},
  // K122 [43,57)  so(3) generators in l=2 rep / sqrt(30)
  {0,0,1, 0.31622777f},{0,1,0,-0.31622777f},{0,2,3,-0.31622777f},{0,3,2, 0.31622777f},
  {0,3,4, 0.18257419f},{0,4,3,-0.18257419f},
  {1,0,4, 0.36514837f},{1,4,0,-0.36514837f},{1,1,3, 0.18257419f},{1,3,1,-0.18257419f},
  {2,1,2, 0.31622777f},{2,2,1,-0.31622777f},{2,0,3,-0.18257419f},{2,3,0, 0.18257419f},
  // K202 [57,62)
  {0,0,0, 0.44721360f},{1,0,1, 0.44721360f},{2,0,2, 0.44721360f},{3,0,3, 0.44721360f},{4,0,4, 0.44721360f},
  // K211 [62,73)  (permutation of K112)
  {0,0,2, 0.31622777f},{0,2,0, 0.31622777f},
  {1,0,1, 0.31622777f},{1,1,0, 0.31622777f},
  {2,0,0,-0.18257419f},{2,1,1, 0.36514837f},{2,2,2,-0.18257419f},
  {3,1,2, 0.31622777f},{3,2,1, 0.31622777f},
  {4,2,2, 0.31622777f},{4,0,0,-0.31622777f},
  // K212 [73,87)  (permutation of K122)
  {0,0,1, 0.31622777f},{1,0,0,-0.31622777f},{2,0,3,-0.31622777f},{3,0,2, 0.31622777f},
  {3,0,4, 0.18257419f},{4,0,3,-0.18257419f},
  {0,1,4, 0.36514837f},{4,1,0,-0.36514837f},{1,1,3, 0.18257419f},{3,1,1,-0.18257419f},
  {1,2,2, 0.31622777f},{2,2,1,-0.31622777f},{0,2,3,-0.18257419f},{3,2,0, 0.18257419f},
  // K220 [87,92)
  {0,0,0, 0.44721360f},{1,1,0, 0.44721360f},{2,2,0, 0.44721360f},{3,3,0, 0.44721360f},{4,4,0, 0.44721360f},
  // K221 [92,106) (permutation of K122)
  {0,1,0, 0.31622777f},{1,0,0,-0.31622777f},{2,3,0,-0.31622777f},{3,2,0, 0.31622777f},
  {3,4,0, 0.18257419f},{4,3,0,-0.18257419f},
  {0,4,1, 0.36514837f},{4,0,1,-0.36514837f},{1,3,1, 0.18257419f},{3,1,1,-0.18257419f},
  {1,2,2, 0.31622777f},{2,1,2,-0.31622777f},{0,3,2,-0.18257419f},{3,0,2, 0.18257419f},
  // K222 [106,120) symmetric l=2 coupling
  {0,1,3, 0.23904572f},{1,0,3, 0.23904572f},{0,3,1, 0.23904572f},{3,0,1, 0.23904572f},
  {1,3,0, 0.23904572f},{3,1,0, 0.23904572f},
  {2,2,2, 0.29880715f},{0,0,2,-0.14940357f},{4,4,2,-0.14940357f},{1,1,2, 0.14940357f},{3,3,2, 0.14940357f},
  {2,0,0, 0.21128856f},{0,2,0, 0.21128856f},{4,2,4, 0.21128856f},
};

// per-instruction metadata, grouped by output l (fan-in concatenation order)
struct ZSeg { int mul, xoff, xstride, yoff, woff, zoff, eb, ee; };
__device__ constexpr ZSeg SEGS[15] = {
  // l = 0  (fan_in 224)
  {128,   0, 1, 0,   0,   0,   0,   1},  // (0,0,0) K000
  { 64, 128, 3, 1, 448, 128,  12,  15},  // (1,1,0) K110
  { 32, 320, 5, 4, 864, 192,  87,  92},  // (2,2,0) K220
  // l = 1  (fan_in 384)
  {128,   0, 1, 1, 128,   0,   1,   4},  // (0,1,1) K011
  { 64, 128, 3, 0, 384, 128,   9,  12},  // (1,0,1) K101
  { 64, 128, 3, 1, 512, 192,  15,  21},  // (1,1,1) K111
  { 64, 128, 3, 4, 640, 256,  32,  43},  // (1,2,1) K121
  { 32, 320, 5, 1, 800, 320,  62,  73},  // (2,1,1) K211
  { 32, 320, 5, 4, 896, 352,  92, 106},  // (2,2,1) K221
  // l = 2  (fan_in 352)
  {128,   0, 1, 4, 256,   0,   4,   9},  // (0,2,2) K022
  { 64, 128, 3, 1, 576, 128,  21,  32},  // (1,1,2) K112
  { 64, 128, 3, 4, 704, 192,  43,  57},  // (1,2,2) K122
  { 32, 320, 5, 0, 768, 256,  57,  62},  // (2,0,2) K202
  { 32, 320, 5, 1, 832, 288,  73,  87},  // (2,1,2) K212
  { 32, 320, 5, 4, 928, 320, 106, 120},  // (2,2,2) K222
};

// ---- per-l worker ----
template<int L, int FANIN, int OUTMUL, int OUTBASE, int SEGB, int SEGE>
__device__ __forceinline__ void process_l(
    char* pool, const float* __restrict__ w, const float* __restrict__ Wg,
    float* __restrict__ out, int n0, int n_edges, int tid, bool full)
{
  constexpr int NK = 2 * L + 1;
  constexpr int ZSTR = FANIN + 4;                  // padded row stride (dwords)
  constexpr int OCOLS = OUTMUL * NK;               // output cols for this l
  constexpr float SQRT2L1 = (L == 0) ? 1.0f : (L == 1) ? 1.7320508f : 2.2360680f;
  constexpr float INVSQRTFAN = (L == 0) ? 0.066815310f : (L == 1) ? 0.051031036f : 0.053300179f;

  const float* sx = (const float*)(pool + OFF_SX);  // [16][480]
  const float* sy = (const float*)(pool + OFF_SY);  // [16][9]
  float* sz   = (float*)(pool + OFF_SZ);            // [NK][16][ZSTR]
  float* sout = (float*)(pool + OFF_SOUT);          // [16][OCOLS]

  const int lane = tid & 31;
  const int wave = tid >> 5;
  const int koff = (lane >> 4) << 1;   // frag K offset: lanes 0-15 -> K0/K1, 16-31 -> K2/K3
  const int mrow = lane & 15;

  // ---- phase 1: all NK z-slices in one sweep; W3J coeffs folded to immediates
  #pragma unroll
  for (int s = SEGB; s < SEGE; ++s) {
    const int mul = SEGS[s].mul;                   // compile-time after unroll
    const int iters = (TILE_E * mul) / 256;
    #pragma unroll
    for (int it = 0; it < iters; ++it) {
      int t = tid + it * 256;
      int e = t / mul;                             // mul is a power of two
      int ul = t - e * mul;
      int n = n0 + e;
      int nc = (n < n_edges) ? n : (n_edges - 1);  // clamp; OOB rows masked at store
      float wv = w[(long)nc * WNUM + SEGS[s].woff + ul] * SQRT2L1;
      const float* xr = sx + e * NDIM1 + SEGS[s].xoff + ul * SEGS[s].xstride;
      const float* yr = sy + e * NDIM2 + SEGS[s].yoff;
      float acc[NK];
      #pragma unroll
      for (int k = 0; k < NK; ++k) acc[k] = 0.0f;
      #pragma unroll
      for (int ei = SEGS[s].eb; ei < SEGS[s].ee; ++ei)
        acc[ENT[ei].c] += ENT[ei].v * xr[ENT[ei].a] * yr[ENT[ei].b];
      #pragma unroll
      for (int k = 0; k < NK; ++k)
        sz[(k * TILE_E + e) * ZSTR + SEGS[s].zoff + ul] = acc[k] * wv;
    }
  }
  __syncthreads();

  // ---- phase 2: (k, col-tile) units spread over all 8 waves; f32 WMMA GEMM
  constexpr int NTILES = OUTMUL / 16;
  constexpr int UNITS = NTILES * NK;
  for (int uu = wave; uu < UNITS; uu += 8) {       // wave-uniform -> EXEC all-1s at WMMA
    int ct = uu / NK;
    int k  = uu - ct * NK;
    const float* Wct = Wg + ct * 16 + mrow;
    const float* az = sz + (k * TILE_E + mrow) * ZSTR;
    v8f acc = {0.f, 0.f, 0.f, 0.f, 0.f, 0.f, 0.f, 0.f};
    for (int kk = 0; kk < FANIN; kk += 4) {
      v2f a, b;
      a[0] = az[kk + koff];                        // A 16x4 f32 layout (ISA 7.12.2)
      a[1] = az[kk + koff + 1];
      b[0] = Wct[(kk + koff) * OUTMUL];            // B 4x16, row K striped across lanes
      b[1] = Wct[(kk + koff + 1) * OUTMUL];
      acc = __builtin_amdgcn_wmma_f32_16x16x4_f32(
          false, a, false, b, (short)0, acc, false, false);
    }
    // D 16x16 f32: VGPR g -> row M = g (lanes 0-15) / g+8 (lanes 16-31), N = lane%16
    int col = (ct * 16 + mrow) * NK + k;
    if (full) {
      // deposit into LDS staging tile; drained below with coalesced async stores
      #pragma unroll
      for (int g = 0; g < 8; ++g) {
        int m = g + ((lane >> 4) << 3);
        sout[m * OCOLS + col] = acc[g] * INVSQRTFAN;
      }
    } else {
      #pragma unroll
      for (int g = 0; g < 8; ++g) {
        int m = g + ((lane >> 4) << 3);
        int n = n0 + m;
        if (n < n_edges) out[(long)n * NDIM1 + OUTBASE + col] = acc[g] * INVSQRTFAN;
      }
    }
  }
  __syncthreads();

  if (full) {
    // ---- drain: LDS -> global, 16B chunks, LDS-direct async stores (ASYNCcnt)
    constexpr int CPR = (OCOLS * 4) / 16;          // 16B chunks per output row
    constexpr int CHUNKS = TILE_E * CPR;
    for (int c = tid; c < CHUNKS; c += 256) {
      int e = c / CPR;
      int off = c - e * CPR;
      unsigned int ldsoff = OFF_SOUT + c * 16;     // sout is row-contiguous
      const char* g = (const char*)(out + (long)(n0 + e) * NDIM1 + OUTBASE) + off * 16;
      asm volatile("global_store_async_from_lds_b128 %0, %1, off"
                   :: "v"(g), "v"(ldsoff) : "memory");
    }
    asm volatile("s_wait_asynccnt 0x0" ::: "memory");
    __syncthreads();                               // sout reusable by next l
  }
}

__global__ __launch_bounds__(256)
void stp_kernel(const float* __restrict__ x, const float* __restrict__ y,
                const float* __restrict__ w,
                const float* __restrict__ W0, const float* __restrict__ W1,
                const float* __restrict__ W2,
                float* __restrict__ out, int n_edges)
{
  extern __shared__ char pool[];
  const int tid = threadIdx.x;
  const int n0 = blockIdx.x * TILE_E;
  const bool full = (n0 + TILE_E <= n_edges);      // always true for N % 16 == 0

  if (full) {
    // async DMA global -> LDS, ASYNCcnt-tracked
    const char* gx = (const char*)(x + (long)n0 * NDIM1);    // 30720 B, 16B-aligned
    for (int c = tid; c < (TILE_E * NDIM1 * 4) / 16; c += 256) {
      unsigned int ldsoff = OFF_SX + c * 16;
      const char* g = gx + c * 16;
      asm volatile("global_load_async_to_lds_b128 %0, %1, off"
                   :: "v"(ldsoff), "v"(g) : "memory");
    }
    if (tid < (TILE_E * NDIM2 * 4) / 16) {                   // 576 B
      unsigned int ldsoff = OFF_SY + tid * 16;
      const char* g = (const char*)(y + (long)n0 * NDIM2) + tid * 16;
      asm volatile("global_load_async_to_lds_b128 %0, %1, off"
                   :: "v"(ldsoff), "v"(g) : "memory");
    }
    asm volatile("s_wait_asynccnt 0x0" ::: "memory");
  } else {
    // ragged tail fallback: guarded scalar staging
    float* sx = (float*)(pool + OFF_SX);
    float* sy = (float*)(pool + OFF_SY);
    for (int idx = tid; idx < TILE_E * NDIM1; idx += 256) {
      int e = idx / NDIM1, c = idx - e * NDIM1;
      int n = n0 + e;
      sx[idx] = (n < n_edges) ? x[(long)n * NDIM1 + c] : 0.0f;
    }
    for (int idx = tid; idx < TILE_E * NDIM2; idx += 256) {
      int e = idx / NDIM2, c = idx - e * NDIM2;
      int n = n0 + e;
      sy[idx] = (n < n_edges) ? y[(long)n * NDIM2 + c] : 0.0f;
    }
  }
  __syncthreads();

  process_l<0, 224, 128,   0,  0,  3>(pool, w, W0, out, n0, n_edges, tid, full);
  process_l<1, 384,  64, 128,  3,  9>(pool, w, W1, out, n0, n_edges, tid, full);
  process_l<2, 352,  32, 320,  9, 15>(pool, w, W2, out, n0, n_edges, tid, full);
}

extern "C" void kernel_launch(void* const* d_in, const int* in_sizes, int n_in,
                              void* d_out, int out_size, void* d_ws, size_t ws_size,
                              hipStream_t stream) {
  const float* x  = (const float*)d_in[0];
  const float* y  = (const float*)d_in[1];
  const float* w  = (const float*)d_in[2];
  const float* W0 = (const float*)d_in[3];
  const float* W1 = (const float*)d_in[4];
  const float* W2 = (const float*)d_in[5];
  float* out = (float*)d_out;
  int n_edges = in_sizes[0] / NDIM1;
  int blocks = (n_edges + TILE_E - 1) / TILE_E;

  // allow >64KB dynamic LDS (CDNA5 WGP has 320KB; ~154KB here -> 2 blocks/WGP)
  (void)hipFuncSetAttribute((const void*)stp_kernel,
                            hipFuncAttributeMaxDynamicSharedMemorySize, LDS_BYTES);
  hipLaunchKernelGGL(stp_kernel, dim3(blocks), dim3(256), LDS_BYTES, stream,
                     x, y, w, W0, W1, W2, out, n_edges);
}